// OpenCVSolver_15118284882649
// MI455X (gfx1250) — compile-verified
//
#include <hip/hip_runtime.h>

typedef float v2f __attribute__((ext_vector_type(2)));
typedef float v8f __attribute__((ext_vector_type(8)));

#define N_ITERS 200
#define LSTR    68          // padded LDS row stride in floats (272B, 16B-aligned rows)

__global__ __launch_bounds__(32, 1)
void sinkhorn_wmma_kernel(const float* __restrict__ Cmat,
                          const float* __restrict__ W1,
                          const float* __restrict__ W2,
                          float* __restrict__ P)
{
    __shared__ float kld[64 * LSTR];              // K (later) / staged C (first), padded
    __shared__ __align__(16) float uld[64];       // a, then final u
    __shared__ __align__(16) float vld[64];       // b, then final v

    const int l  = threadIdx.x;       // 0..31 (wave32)
    const int hl = l >> 4;            // 0/1 half-wave
    const int cl = l & 15;
    const int b  = blockIdx.x;

    // ---------------- async-stage this batch's 16KB C tile into padded LDS ----------------
    {
        unsigned ldsa = (unsigned)(uintptr_t)&kld[hl * LSTR + 4 * cl];
        const float* g = Cmat + (size_t)b * 4096 + 4 * l;   // 16B per lane
#pragma unroll
        for (int it = 0; it < 32; ++it) {
            asm volatile("global_load_async_to_lds_b128 %0, %1, off"
                         :: "v"(ldsa), "v"(g) : "memory");
            ldsa += 2 * LSTR * 4;   // two rows per step (padded)
            g    += 128;            // 512B per step (unpadded global)
        }
    }

    // ---------------- marginals a,b while the async DMA flies ----------------
    {
        float x0 = W1[(size_t)b * 64 + l];
        float x1 = W1[(size_t)b * 64 + 32 + l];
        x0 = fmaxf(x0, 0.0f) + 1e-5f;
        x1 = fmaxf(x1, 0.0f) + 1e-5f;
        float s = x0 + x1;
        for (int o = 16; o; o >>= 1) s += __shfl_xor(s, o, 32);
        float sc = 64.0f / s;
        uld[l] = x0 * sc; uld[32 + l] = x1 * sc;

        float y0 = W2[(size_t)b * 64 + l];
        float y1 = W2[(size_t)b * 64 + 32 + l];
        y0 = fmaxf(y0, 0.0f) + 1e-5f;
        y1 = fmaxf(y1, 0.0f) + 1e-5f;
        float t = y0 + y1;
        for (int o = 16; o; o >>= 1) t += __shfl_xor(t, o, 32);
        float tc = 64.0f / t;
        vld[l] = y0 * tc; vld[32 + l] = y1 * tc;
    }
    // lane-distributed marginals: element 16t+cl in lanes cl and cl+16
    float Aa[4], Bb[4];
#pragma unroll
    for (int t = 0; t < 4; ++t) { Aa[t] = uld[16 * t + cl]; Bb[t] = vld[16 * t + cl]; }

    // ---------------- wait for C, build K = exp(-C/eps) in place ----------------
    asm volatile("s_wait_asynccnt 0x0" ::: "memory");
#pragma unroll
    for (int it = 0; it < 32; ++it) {
        float4* p = (float4*)&kld[(2 * it + hl) * LSTR + 4 * cl];
        float4 c4 = *p;
        c4.x = __expf(-20.0f * c4.x);
        c4.y = __expf(-20.0f * c4.y);
        c4.z = __expf(-20.0f * c4.z);
        c4.w = __expf(-20.0f * c4.w);
        *p = c4;
    }

    // ---------------- pack K and K^T into WMMA B-operand register tiles ----------------
    // B-tile (chunk c = k rows 4c..4c+3, tile t = cols 16t..16t+15), assumed layout:
    //   vgpr0: lanes0-15 -> row 4c+0, lanes16-31 -> row 4c+2 ; vgpr1: rows 4c+1 / 4c+3
    v2f BK[16][4];    // B-layout of K    (for z = K^T u : D = u_rep x K)
    v2f BKT[16][4];   // B-layout of K^T  (for y = K v   : D = v_rep x K^T)
    {
        const float* kb  = &kld[(2 * hl) * LSTR + cl];   // K [4c+2hl][16t+cl]
        const float* ktb = &kld[cl * LSTR + 2 * hl];     // K^T[4c+2hl][16t+cl] = K[16t+cl][4c+2hl]
#pragma unroll
        for (int c = 0; c < 16; ++c)
#pragma unroll
            for (int t = 0; t < 4; ++t) {
                v2f r; r.x = kb[(4 * c) * LSTR + 16 * t];
                       r.y = kb[(4 * c + 1) * LSTR + 16 * t];
                BK[c][t] = r;
                v2f q; q.x = ktb[(16 * t) * LSTR + 4 * c];
                       q.y = ktb[(16 * t) * LSTR + 4 * c + 1];
                BKT[c][t] = q;
            }
    }

    // ---------------- Sinkhorn main loop: u = a/(Kv), v = b/(K^T u) ----------------
    float U[4];
    float V[4] = {1.0f, 1.0f, 1.0f, 1.0f};
    const int pa0 = 8 * hl;                 // bpermute byte offset: A vgpr0 wants v[4c+2*hl]
    const v8f Z = {0, 0, 0, 0, 0, 0, 0, 0};

#pragma unroll 1
    for (int itn = 0; itn < N_ITERS; ++itn) {
        // ---- y = K v  (A = v replicated across rows, B = K^T tiles) ----
        v8f D[4]; D[0] = Z; D[1] = Z; D[2] = Z; D[3] = Z;
#pragma unroll
        for (int c = 0; c < 16; ++c) {
            const int src = __float_as_int(V[c >> 2]);
            const int base = ((4 * c) & 15) * 4;
            v2f Av;
            Av.x = __int_as_float(__builtin_amdgcn_ds_bpermute(pa0 + base,     src));
            Av.y = __int_as_float(__builtin_amdgcn_ds_bpermute(pa0 + base + 4, src));
#pragma unroll
            for (int t = 0; t < 4; ++t)
                D[t] = __builtin_amdgcn_wmma_f32_16x16x4_f32(
                           false, Av, false, BKT[c][t], (short)0, D[t], false, false);
        }
#pragma unroll
        for (int t = 0; t < 4; ++t)
            U[t] = Aa[t] * __builtin_amdgcn_rcpf(D[t][0]);   // u[16t+cl], lane-distributed

        // ---- z = K^T u  (A = u replicated, B = K tiles) ----
        D[0] = Z; D[1] = Z; D[2] = Z; D[3] = Z;
#pragma unroll
        for (int c = 0; c < 16; ++c) {
            const int src = __float_as_int(U[c >> 2]);
            const int base = ((4 * c) & 15) * 4;
            v2f Av;
            Av.x = __int_as_float(__builtin_amdgcn_ds_bpermute(pa0 + base,     src));
            Av.y = __int_as_float(__builtin_amdgcn_ds_bpermute(pa0 + base + 4, src));
#pragma unroll
            for (int t = 0; t < 4; ++t)
                D[t] = __builtin_amdgcn_wmma_f32_16x16x4_f32(
                           false, Av, false, BK[c][t], (short)0, D[t], false, false);
        }
#pragma unroll
        for (int t = 0; t < 4; ++t)
            V[t] = Bb[t] * __builtin_amdgcn_rcpf(D[t][0]);
    }

    // ---------------- epilogue: P = u .* K .* v^T, coalesced b128 stores ----------------
#pragma unroll
    for (int t = 0; t < 4; ++t) { uld[16 * t + cl] = U[t]; vld[16 * t + cl] = V[t]; }

    float* out = P + (size_t)b * 4096;
    const float4 v4 = *(const float4*)&vld[4 * cl];
#pragma unroll
    for (int it = 0; it < 32; ++it) {
        const int row = 2 * it + hl;
        const float ui = uld[row];
        const float4 k4 = *(const float4*)&kld[row * LSTR + 4 * cl];
        float4 o;
        o.x = ui * k4.x * v4.x;
        o.y = ui * k4.y * v4.y;
        o.z = ui * k4.z * v4.z;
        o.w = ui * k4.w * v4.w;
        *(float4*)(out + row * 64 + 4 * cl) = o;
    }
}

extern "C" void kernel_launch(void* const* d_in, const int* in_sizes, int n_in,
                              void* d_out, int out_size, void* d_ws, size_t ws_size,
                              hipStream_t stream) {
    const float* C  = (const float*)d_in[0];   // [B,1,64,64]
    const float* W1 = (const float*)d_in[1];   // [B,1,64]
    const float* W2 = (const float*)d_in[2];   // [B,1,64]
    float* P = (float*)d_out;                  // [B,64,64]
    const int B = in_sizes[1] / 64;            // 8192
    sinkhorn_wmma_kernel<<<dim3(B), dim3(32), 0, stream>>>(C, W1, W2, P);
}